// TorStep_64458869179068
// MI455X (gfx1250) — compile-verified
//
#include <hip/hip_runtime.h>

// ---------------------------------------------------------------------------
// TOR step for MI455X (gfx1250, wave32).
//   - one wave per LOR: 32 lanes x 4 samples = 128 samples
//   - trilinear gather from z-space image with per-view permuted strides
//   - exact f32 wave reduction via V_WMMA_F32_16X16X4_F32 (A = ones)
//   - fused backprojection scatter (global_atomic_add_f32) into d_out
//   - finalize: out = img * eff * out
// ---------------------------------------------------------------------------

namespace {
constexpr int   kNLors        = 50000;
constexpr int   kNSamples     = 128;
constexpr int   kG            = 160;
constexpr int   kNVox         = kG * kG * kG;          // 4,096,000
constexpr float kKW           = 3.0f;                  // KERNEL_WIDTH
constexpr int   kWavesPerBlk  = 8;                     // 256 threads
constexpr float kInvSamples   = 1.0f / (float)kNSamples;
// f = (pos - origin)/vsize - 0.5 with origin=-100, vsize=1.25:
//   f = pos*0.8 + 79.5
constexpr float kScale        = 0.8f;
constexpr float kBias         = 79.5f;
}  // namespace

typedef __attribute__((ext_vector_type(2))) float v2f;
typedef __attribute__((ext_vector_type(8))) float v8f;

// Sum 128 values (4 per lane, wave32) exactly in f32.
// Two V_WMMA_F32_16X16X4_F32 with A = ones(16x4): D[m,n] = sum_k B[k,n].
// Every row of D is identical, so c[0] on lane L holds colsum[L%16].
// Summing all 16 column sums == summing every B entry, independent of the
// exact lane->B placement, so this is layout-robust and bit-exact f32.
__device__ __forceinline__ float wave_sum128(float v0, float v1, float v2, float v3) {
    v2f a;   a[0]   = 1.0f; a[1]   = 1.0f;   // A = ones
    v2f b01; b01[0] = v0;   b01[1] = v1;
    v2f b23; b23[0] = v2;   b23[1] = v3;
    v8f c = {};
    c = __builtin_amdgcn_wmma_f32_16x16x4_f32(false, a, false, b01, (short)0, c, false, false);
    c = __builtin_amdgcn_wmma_f32_16x16x4_f32(false, a, false, b23, (short)0, c, false, false);
    float s = c[0];
    s += __shfl_xor(s, 1, 32);
    s += __shfl_xor(s, 2, 32);
    s += __shfl_xor(s, 4, 32);
    s += __shfl_xor(s, 8, 32);
    return s;  // total over all 128 samples, replicated in every lane
}

// SA/SB/SC: z-space strides for view dims 0/1/2 (compile-time so that the
// 8 corner addresses become constant offsets and the stride-1 pair merges
// into a single global_load_b64).
template <int SA, int SB, int SC>
__global__ __launch_bounds__(256)
void tor_axis_kernel(const float* __restrict__ lors,   // [6, kNLors]
                     const float* __restrict__ img,    // z-space image
                     float* __restrict__ bp)           // z-space accumulator
{
    const int wave = threadIdx.x >> 5;
    const int lane = threadIdx.x & 31;
    const int lor  = blockIdx.x * kWavesPerBlk + wave;
    if (lor >= kNLors) return;  // wave-uniform (never taken: 6250*8 == 50000)

    const float p1x = lors[0 * kNLors + lor];
    const float p1y = lors[1 * kNLors + lor];
    const float p1z = lors[2 * kNLors + lor];
    const float ddx = lors[3 * kNLors + lor] - p1x;
    const float ddy = lors[4 * kNLors + lor] - p1y;
    const float ddz = lors[5 * kNLors + lor] - p1z;
    const float step = sqrtf(ddx * ddx + ddy * ddy + ddz * ddz) * kInvSamples;

    int   ix[4], iy[4], iz[4];
    float wxa[4], wya[4], wza[4], val[4];

#pragma unroll
    for (int j = 0; j < 4; ++j) {
        const float t = ((float)(lane * 4 + j) + 0.5f) * kInvSamples;
        float fx = fmaf(fmaf(t, ddx, p1x), kScale, kBias);
        float fy = fmaf(fmaf(t, ddy, p1y), kScale, kBias);
        float fz = fmaf(fmaf(t, ddz, p1z), kScale, kBias);
        fx = fminf(fmaxf(fx, 0.0f), 159.0f);
        fy = fminf(fmaxf(fy, 0.0f), 159.0f);
        fz = fminf(fmaxf(fz, 0.0f), 159.0f);
        const float gx = fminf(floorf(fx), 158.0f);
        const float gy = fminf(floorf(fy), 158.0f);
        const float gz = fminf(floorf(fz), 158.0f);
        ix[j] = (int)gx; iy[j] = (int)gy; iz[j] = (int)gz;
        wxa[j] = fx - gx; wya[j] = fy - gy; wza[j] = fz - gz;

        const float* pb = img + ix[j] * SA + iy[j] * SB + iz[j] * SC;
        const float x0 = 1.0f - wxa[j], x1 = wxa[j];
        const float y0 = 1.0f - wya[j], y1 = wya[j];
        const float z0 = 1.0f - wza[j], z1 = wza[j];
        val[j] = x0 * (y0 * (z0 * pb[0]       + z1 * pb[SC])
                     + y1 * (z0 * pb[SB]      + z1 * pb[SB + SC]))
               + x1 * (y0 * (z0 * pb[SA]      + z1 * pb[SA + SC])
                     + y1 * (z0 * pb[SA + SB] + z1 * pb[SA + SB + SC]));
    }

    // line integral p = KW*step*total; scatter weight = KW*p*step
    const float total   = wave_sum128(val[0], val[1], val[2], val[3]);
    const float contrib = (kKW * kKW) * step * step * total;

#pragma unroll
    for (int j = 0; j < 4; ++j) {
        float* pb = bp + ix[j] * SA + iy[j] * SB + iz[j] * SC;
        const float x0 = (1.0f - wxa[j]) * contrib, x1 = wxa[j] * contrib;
        const float y0 = 1.0f - wya[j], y1 = wya[j];
        const float z0 = 1.0f - wza[j], z1 = wza[j];
        atomicAdd(pb + 0,            x0 * y0 * z0);
        atomicAdd(pb + SC,           x0 * y0 * z1);
        atomicAdd(pb + SB,           x0 * y1 * z0);
        atomicAdd(pb + SB + SC,      x0 * y1 * z1);
        atomicAdd(pb + SA,           x1 * y0 * z0);
        atomicAdd(pb + SA + SC,      x1 * y0 * z1);
        atomicAdd(pb + SA + SB,      x1 * y1 * z0);
        atomicAdd(pb + SA + SB + SC, x1 * y1 * z1);
    }
}

__global__ void zero_kernel(float4* __restrict__ out) {
    const int i = blockIdx.x * blockDim.x + threadIdx.x;
    out[i] = make_float4(0.0f, 0.0f, 0.0f, 0.0f);
}

__global__ void finalize_kernel(const float4* __restrict__ img,
                                const float4* __restrict__ eff,
                                float4* __restrict__ out) {
    const int i = blockIdx.x * blockDim.x + threadIdx.x;
    float4 o = out[i];
    const float4 a = img[i];
    const float4 b = eff[i];
    o.x *= a.x * b.x;
    o.y *= a.y * b.y;
    o.z *= a.z * b.z;
    o.w *= a.w * b.w;
    out[i] = o;
}

extern "C" void kernel_launch(void* const* d_in, const int* in_sizes, int n_in,
                              void* d_out, int out_size, void* d_ws, size_t ws_size,
                              hipStream_t stream) {
    (void)in_sizes; (void)n_in; (void)out_size; (void)d_ws; (void)ws_size;
    const float* img = (const float*)d_in[0];
    const float* eff = (const float*)d_in[1];
    const float* xl  = (const float*)d_in[2];   // [6, 50000]
    const float* yl  = (const float*)d_in[3];
    const float* zl  = (const float*)d_in[4];
    float* out = (float*)d_out;

    const int vecN   = kNVox / 4;               // 1,024,000 (exact)
    const int vecBlk = vecN / 256;              // 4000 (exact)
    zero_kernel<<<vecBlk, 256, 0, stream>>>((float4*)out);

    const int blocks = kNLors / kWavesPerBlk;   // 6250 (exact)
    // z-view: imgz[v0,v1,v2]            -> strides (25600, 160, 1)
    tor_axis_kernel<25600, 160, 1><<<blocks, 256, 0, stream>>>(zl, img, out);
    // x-view: imgx[a,b,c] = imgz[b,c,a] -> strides (1, 25600, 160)
    tor_axis_kernel<1, 25600, 160><<<blocks, 256, 0, stream>>>(xl, img, out);
    // y-view: imgy[a,b,c] = imgz[b,a,c] -> strides (160, 25600, 1)
    tor_axis_kernel<160, 25600, 1><<<blocks, 256, 0, stream>>>(yl, img, out);

    finalize_kernel<<<vecBlk, 256, 0, stream>>>((const float4*)img,
                                                (const float4*)eff,
                                                (float4*)out);
}